// DiagonalBandAttention_18786186953351
// MI455X (gfx1250) — compile-verified
//
#include <hip/hip_runtime.h>

// Problem constants (from reference setup_inputs)
#define BB   2
#define CC   192
#define SS   512
#define BW   21      // band width
#define HALF 10      // BW/2
#define KS   7       // depthwise kernel
#define PLANES (BB*CC)          // 384
#define ROWLEN SS               // 512

typedef float v2f __attribute__((ext_vector_type(2)));
typedef float v8f __attribute__((ext_vector_type(8)));

// -------------------------------------------------------------------------
// Kernel 1: band[b,c,j] = (1/21) * sum_{k=-10..10} x[b,c, j+k, j]
// LDS-tiled so global reads are coalesced row segments instead of
// stride-(S+1) diagonal gathers. Tile: 64 columns, 84 rows.
// LDS column read tile[(j+k)*64 + j]: bank = (65j + 64k) % 64 = j -> no
// bank conflicts across the 64 computing lanes.
// -------------------------------------------------------------------------
__global__ void band_mean_kernel(const float* __restrict__ x,
                                 float* __restrict__ band) {
    __shared__ float tile[84 * 64];
    int bid     = blockIdx.x;        // plane*8 + tileIdx
    int tileIdx = bid & 7;           // 8 tiles of 64 cols per plane
    int plane   = bid >> 3;          // 0..383
    int c0      = tileIdx * 64;
    const float* xp = x + (long)plane * SS * SS;

    for (int idx = threadIdx.x; idx < 84 * 64; idx += blockDim.x) {
        int r  = idx >> 6;           // 0..83
        int cj = idx & 63;
        int g  = c0 - HALF + r;      // global row
        tile[idx] = (g >= 0 && g < SS) ? xp[(long)g * SS + c0 + cj] : 0.0f;
    }
    __syncthreads();

    int t = threadIdx.x;
    if (t < 64) {
        float s = 0.0f;
        #pragma unroll
        for (int k = 0; k < BW; ++k)
            s += tile[(t + k) * 64 + t];
        band[plane * SS + c0 + t] = s * (1.0f / 21.0f);
    }
}

// -------------------------------------------------------------------------
// Kernel 2: depthwise conv1d, k=7, pad=3, groups=C, no bias (cross-corr.)
// -------------------------------------------------------------------------
__global__ void dwconv_kernel(const float* __restrict__ band,
                              const float* __restrict__ w,   // [C,1,7]
                              float* __restrict__ attn1) {
    int gid = blockIdx.x * blockDim.x + threadIdx.x;   // B*C*S threads
    if (gid >= PLANES * SS) return;
    int s = gid & (SS - 1);
    int c = (gid >> 9) % CC;
    const float* bp = band + (gid - s);
    const float* wp = w + c * KS;
    float acc = 0.0f;
    #pragma unroll
    for (int t = 0; t < KS; ++t) {
        int ss = s + t - 3;
        if (ss >= 0 && ss < SS) acc += bp[ss] * wp[t];
    }
    attn1[gid] = acc;
}

// -------------------------------------------------------------------------
// Kernel 3: pointwise conv as WMMA f32 GEMM + bias.
//   attn2[b,d,s] = sum_c pw[d,c] * attn1[b,c,s] + pb[d]
// One wave per 16x16 output tile; K-loop of 48 v_wmma_f32_16x16x4_f32.
// A 16x4 layout : lane m (<16) holds {K0,K1}; lane m+16 holds {K2,K3}.
// B 4x16 layout : lane n (<16) holds {K0,K1}; lane n+16 holds {K2,K3}.
// D 16x16 layout: VGPR v, lanes 0-15 -> M=v, N=lane; lanes 16-31 -> M=v+8.
// -------------------------------------------------------------------------
__global__ void pointwise_wmma_kernel(const float* __restrict__ pw,   // [C,C]
                                      const float* __restrict__ attn1,// [B,C,S]
                                      const float* __restrict__ pb,   // [C]
                                      float* __restrict__ attn2) {    // [B,C,S]
    const int MT = CC / 16;            // 12 tiles in d
    const int NT = SS / 16;            // 32 tiles in s
    int wave   = (blockIdx.x * blockDim.x + threadIdx.x) >> 5;  // global wave id
    int lane   = threadIdx.x & 31;
    int tileId = wave;                 // 0 .. B*MT*NT-1 = 767
    int b   = tileId / (MT * NT);
    int rem = tileId % (MT * NT);
    int d0  = (rem / NT) * 16;
    int s0  = (rem % NT) * 16;

    int half = lane >> 4;              // 0: K 0,1   1: K 2,3
    int l    = lane & 15;              // row (A) / col (B,D) index

    v8f acc = {};
    for (int kk = 0; kk < CC; kk += 4) {
        v2f av, bv;
        const float* ap = pw + (d0 + l) * CC + kk + 2 * half;
        av.x = ap[0];
        av.y = ap[1];
        const float* bp = attn1 + ((long)(b * CC + kk + 2 * half)) * SS + s0 + l;
        bv.x = bp[0];
        bv.y = bp[SS];
        acc = __builtin_amdgcn_wmma_f32_16x16x4_f32(
            /*neg_a=*/false, av, /*neg_b=*/false, bv,
            /*c_mod=*/(short)0, acc, /*reuse_a=*/false, /*reuse_b=*/false);
    }
    #pragma unroll
    for (int v = 0; v < 8; ++v) {
        int m = v + 8 * half;
        attn2[((long)(b * CC + d0 + m)) * SS + s0 + l] = acc[v] + pb[d0 + m];
    }
}

// -------------------------------------------------------------------------
// Kernel 4: softmax over s per (b,d) row, in place. 512 threads per row.
// -------------------------------------------------------------------------
__global__ void softmax_kernel(float* __restrict__ attn2) {
    __shared__ float red[SS];
    int row = blockIdx.x;              // 0..383
    float* p = attn2 + (long)row * SS;
    int t = threadIdx.x;               // 0..511
    float v = p[t];
    red[t] = v;
    __syncthreads();
    for (int s = SS / 2; s > 0; s >>= 1) {
        if (t < s) red[t] = fmaxf(red[t], red[t + s]);
        __syncthreads();
    }
    float m = red[0];
    __syncthreads();
    float e = __expf(v - m);
    red[t] = e;
    __syncthreads();
    for (int s = SS / 2; s > 0; s >>= 1) {
        if (t < s) red[t] += red[t + s];
        __syncthreads();
    }
    p[t] = e / red[0];
}

// -------------------------------------------------------------------------
// Kernel 5: out = x with main diagonal scaled by attn. float4 streaming copy
// (the only kernel that matters: ~805 MB -> ~35 us at 23.3 TB/s).
// -------------------------------------------------------------------------
__global__ void scale_copy_kernel(const float* __restrict__ x,
                                  const float* __restrict__ attn,
                                  float* __restrict__ out) {
    long i = (long)blockIdx.x * blockDim.x + threadIdx.x;   // float4 index
    float4 v = ((const float4*)x)[i];
    long e    = i << 2;
    int plane = (int)(e >> 18);          // SS*SS = 2^18
    int off   = (int)(e & ((SS * SS) - 1));
    int row   = off >> 9;
    int cb    = off & (SS - 1);          // first column of this float4
    if (row >= cb && row <= cb + 3) {    // diagonal element inside this quad
        float a = attn[plane * SS + row];
        ((float*)&v)[row - cb] *= a;
    }
    ((float4*)out)[i] = v;
}

// -------------------------------------------------------------------------
extern "C" void kernel_launch(void* const* d_in, const int* in_sizes, int n_in,
                              void* d_out, int out_size, void* d_ws, size_t ws_size,
                              hipStream_t stream) {
    const float* x      = (const float*)d_in[0];   // [2,192,512,512]
    const float* conv_w = (const float*)d_in[1];   // [192,1,7]
    const float* pw     = (const float*)d_in[2];   // [192,192]
    const float* pb     = (const float*)d_in[3];   // [192]
    float* out          = (float*)d_out;

    float* ws    = (float*)d_ws;
    float* band  = ws;                       // 196608 floats
    float* attn1 = ws + PLANES * SS;         // 196608 floats
    float* attn2 = ws + 2 * PLANES * SS;     // 196608 floats

    // 1) band mean: 384 planes x 8 column-tiles
    band_mean_kernel<<<PLANES * 8, 128, 0, stream>>>(x, band);

    // 2) depthwise conv: 196608 threads
    dwconv_kernel<<<(PLANES * SS) / 256, 256, 0, stream>>>(band, conv_w, attn1);

    // 3) pointwise GEMM via WMMA: 768 wave-tiles, 4 waves/block
    pointwise_wmma_kernel<<<(BB * (CC / 16) * (SS / 16)) / 4, 128, 0, stream>>>(
        pw, attn1, pb, attn2);

    // 4) softmax per (b,d) row, in place
    softmax_kernel<<<PLANES, SS, 0, stream>>>(attn2);

    // 5) streaming copy with diagonal scale: 25165824 float4s
    scale_copy_kernel<<<(PLANES * SS * SS / 4) / 256, 256, 0, stream>>>(x, attn2, out);
}